// Generator_78881369358886
// MI455X (gfx1250) — compile-verified
//
#include <hip/hip_runtime.h>

// MI455X (gfx1250) — wave32, WMMA. fp32 path via V_WMMA_F32_16X16X4_F32.

typedef __attribute__((ext_vector_type(2))) float v2f;
typedef __attribute__((ext_vector_type(8))) float v8f;

#define IN_DIM 310
#define H1_DIM 128
#define D_DIM  100
#define SLOPE  0.01f

__device__ __forceinline__ float leaky(float v) {
    return v >= 0.0f ? v : SLOPE * v;
}

// D = A(16x4,f32) * B(4x16,f32) + C(16x16,f32)
__device__ __forceinline__ v8f wmma4(v2f a, v2f b, v8f c) {
    return __builtin_amdgcn_wmma_f32_16x16x4_f32(
        /*neg_a=*/false, a, /*neg_b=*/false, b,
        /*c_mod=*/(short)0, c, /*reuse_a=*/false, /*reuse_b=*/false);
}

__global__ __launch_bounds__(128, 1)
void mlp_poly_kernel(const float* __restrict__ x,
                     const float* __restrict__ data_t,
                     const float* __restrict__ W1, const float* __restrict__ b1,
                     const float* __restrict__ W2, const float* __restrict__ b2,
                     const float* __restrict__ W3, const float* __restrict__ b3,
                     float* __restrict__ out)
{
    // Block = 4 waves, each wave owns a 16-row batch tile (64 rows/block).
    // Weight/activation LDS layouts keep every v2f WMMA fragment contiguous
    // and 8B-aligned so fragments load as single ds_load_b64 into a VGPR pair.
    alignas(16) __shared__ float sW2p[6][H1_DIM][2];  // [kpair][col][k&1]; rows 10,11 zero
    alignas(16) __shared__ float sB2[H1_DIM];
    alignas(16) __shared__ float sW3p[64][16][2];     // [kpair][col][k&1]; cols 2..15 zero
    alignas(16) __shared__ float sHB[4][16][18];      // per-wave h1 tile (even stride)
    alignas(16) __shared__ float sH2[4][16][130];     // per-wave h2 tile (even stride)
    alignas(16) __shared__ float sCO[4][16][2];       // per-wave coeff

    const int tid  = threadIdx.x;
    const int wave = tid >> 5;
    const int lane = tid & 31;
    const int half = lane >> 4;            // 0: lanes 0-15, 1: lanes 16-31
    const int n    = lane & 15;            // M for A-frag, N for B/C/D frags
    const int kh   = half * 2;             // K sub-offset carried by lane half

    // ---- stage packed/padded weights into LDS (once per block) ----
    for (int i = tid; i < 12 * H1_DIM; i += 128) {
        int r = i >> 7, c = i & 127;
        sW2p[r >> 1][c][r & 1] = (r < 10) ? W2[r * H1_DIM + c] : 0.0f;
    }
    sB2[tid] = b2[tid];                    // blockDim == 128 == H1_DIM
    for (int i = tid; i < H1_DIM * 16; i += 128) {
        int r = i >> 4, c = i & 15;
        sW3p[r >> 1][c][r & 1] = (c < 2) ? W3[r * 2 + c] : 0.0f;
    }
    __syncthreads();

    const int row0 = blockIdx.x * 64 + wave * 16;

    // Early prefetch of this wave's data_t tile (16 rows x 100 f32 = 6400 B)
    const float* tbase = data_t + (size_t)row0 * D_DIM;
    __builtin_prefetch(tbase + lane * 32, 0, 1);
    if (lane < 18) __builtin_prefetch(tbase + (32 + lane) * 32, 0, 1);

    // =========== Layer 1: h1 = leaky(x @ W1 + b1), [16 x 10] ===========
    // A-frag: lane holds row (row0+n), K = k + kh + {0,1}  -> 8B contiguous loads
    // B-frag: lane holds W1 column nc; pad lanes (n>=10) duplicate column 9 —
    // harmless since D columns 10..15 are zeroed at the h1 store. All loads are
    // unconditional + in-bounds -> no EXEC save/restore in the hot loop.
    const int nc = (n < 10) ? n : 9;
    const float* w1p  = W1 + nc;
    const float* xrow = x + (size_t)(row0 + n) * IN_DIM + kh;

    v8f accA = {};
    v8f accB = {};
    for (int k = 0; k + 8 <= 308; k += 8) {      // two independent WMMA chains
        const int kr = k + kh;
        v2f a0, b0, a1, b1v;
        a0.x = xrow[k];     a0.y = xrow[k + 1];
        a1.x = xrow[k + 4]; a1.y = xrow[k + 5];
        b0.x  = w1p[(kr    ) * 10];
        b0.y  = w1p[(kr + 1) * 10];
        b1v.x = w1p[(kr + 4) * 10];
        b1v.y = w1p[(kr + 5) * 10];
        accA = wmma4(a0, b0,  accA);
        accB = wmma4(a1, b1v, accB);
    }
    {   // tail: K = 308..309 (half-1 lanes carry K=310,311 -> zero).
        // Safe-pointer select keeps all loads in-bounds; value select is a cndmask.
        const bool tv = (kh == 0);
        const float* xs = xrow + (tv ? 308 : 0);
        const float* ws = w1p  + (tv ? 3080 : 0);   // 308 * 10
        float xa0 = xs[0], xa1 = xs[1];
        float wb0 = ws[0], wb1 = ws[10];
        v2f a, bv;
        a.x  = tv ? xa0 : 0.0f;
        a.y  = tv ? xa1 : 0.0f;
        bv.x = tv ? wb0 : 0.0f;
        bv.y = tv ? wb1 : 0.0f;
        accA = wmma4(a, bv, accA);
    }
    v8f acc1 = accA + accB;

    // bias + leaky, park h1 tile in LDS (zero the padded cols 10..15)
    {
        float bias1 = b1[nc];                     // unconditional, in-bounds
        #pragma unroll
        for (int v = 0; v < 8; ++v) {
            int m = v + 8 * half;                 // D row
            float val = leaky(acc1[v] + bias1);
            sHB[wave][m][n] = (n < 10) ? val : 0.0f;
        }
    }

    // =========== Layer 2: h2 = leaky(h1 @ W2 + b2), [16 x 128] ===========
    {
        #pragma unroll
        for (int t = 0; t < 8; ++t) {             // 8 output tiles of N=16
            v8f acc = {};
            #pragma unroll
            for (int k = 0; k < 12; k += 4) {     // K padded 10 -> 12
                v2f a  = *(const v2f*)&sHB[wave][n][k + kh];          // ds_load_b64
                v2f bv = *(const v2f*)&sW2p[(k >> 1) + half][t * 16 + n][0];
                acc = wmma4(a, bv, acc);
            }
            float bias2 = sB2[t * 16 + n];
            #pragma unroll
            for (int v = 0; v < 8; ++v) {
                sH2[wave][v + 8 * half][t * 16 + n] = leaky(acc[v] + bias2);
            }
        }
    }

    // =========== Layer 3: coeff = leaky(h2 @ W3 + b3), [16 x 2] ===========
    {
        v8f accC = {};
        v8f accD = {};
        for (int k = 0; k < 128; k += 8) {        // two chains again
            v2f a0  = *(const v2f*)&sH2[wave][n][k + kh];
            v2f a1  = *(const v2f*)&sH2[wave][n][k + kh + 4];
            v2f b0  = *(const v2f*)&sW3p[(k >> 1) + half][n][0];
            v2f b1v = *(const v2f*)&sW3p[(k >> 1) + 2 + half][n][0];
            accC = wmma4(a0, b0,  accC);
            accD = wmma4(a1, b1v, accD);
        }
        v8f acc3 = accC + accD;
        float bias3 = b3[n & 1];                  // unconditional, in-bounds
        #pragma unroll
        for (int v = 0; v < 8; ++v) {
            float val = leaky(acc3[v] + bias3);
            if (n < 2) sCO[wave][v + 8 * half][n] = val;
        }
    }

    // =========== out[r, d] = c0*t + c1*t^2 = t * fma(c1, t, c0) ===========
    for (int r = 0; r < 16; ++r) {
        float c0 = sCO[wave][r][0];               // LDS broadcast
        float c1 = sCO[wave][r][1];
        const float* trow = data_t + (size_t)(row0 + r) * D_DIM;
        float*       orow = out    + (size_t)(row0 + r) * D_DIM;
        #pragma unroll
        for (int d = lane; d < D_DIM; d += 32) {  // coalesced
            float t = trow[d];
            orow[d] = t * fmaf(c1, t, c0);
        }
    }
}

extern "C" void kernel_launch(void* const* d_in, const int* in_sizes, int n_in,
                              void* d_out, int out_size, void* d_ws, size_t ws_size,
                              hipStream_t stream) {
    (void)n_in; (void)out_size; (void)d_ws; (void)ws_size;
    const float* x  = (const float*)d_in[0];
    const float* dt = (const float*)d_in[1];
    const float* W1 = (const float*)d_in[2];
    const float* b1 = (const float*)d_in[3];
    const float* W2 = (const float*)d_in[4];
    const float* b2 = (const float*)d_in[5];
    const float* W3 = (const float*)d_in[6];
    const float* b3 = (const float*)d_in[7];
    float* out = (float*)d_out;

    const int nrows = in_sizes[0] / IN_DIM;       // 131072
    const int grid  = nrows / 64;                 // 64 rows per 128-thread block
    mlp_poly_kernel<<<grid, 128, 0, stream>>>(x, dt, W1, b1, W2, b2, W3, b3, out);
}